// _SparseChainPGO_29351806500962
// MI455X (gfx1250) — compile-verified
//
#include <hip/hip_runtime.h>
#include <math.h>

#define BLOCK 256
#define WAVES (BLOCK / 32)
#define EPS_F 1e-6f

// ---- CDNA5 async global<->LDS path detection (device pass only) ------------
#if defined(__HIP_DEVICE_COMPILE__) && defined(__has_builtin)
#  if __has_builtin(__builtin_amdgcn_global_load_async_to_lds_b32)
#    define ASYNC_LDS 1
#  endif
#  if __has_builtin(__builtin_amdgcn_global_store_async_from_lds_b32)
#    define ASYNC_ST 1
#  endif
#endif
#ifndef ASYNC_LDS
#  define ASYNC_LDS 0
#endif
#ifndef ASYNC_ST
#  define ASYNC_ST 0
#endif

#if ASYNC_LDS || ASYNC_ST
typedef __attribute__((address_space(1))) int gas_i32; // global AS
typedef __attribute__((address_space(3))) int las_i32; // LDS AS

__device__ __forceinline__ void wait_async0() {
#if __has_builtin(__builtin_amdgcn_s_wait_asynccnt)
  __builtin_amdgcn_s_wait_asynccnt(0);
#else
  asm volatile("s_wait_asynccnt 0" ::: "memory");
#endif
}
__device__ __forceinline__ void wait_ds0() {
#if __has_builtin(__builtin_amdgcn_s_wait_dscnt)
  __builtin_amdgcn_s_wait_dscnt(0);
#else
  asm volatile("s_wait_dscnt 0" ::: "memory");
#endif
}
#endif

// ---- tiny SE(3) helpers (fp32, matching the reference math) ----------------
struct V3 { float x, y, z; };
struct Q4 { float x, y, z, w; };
struct SE3 { V3 t; Q4 q; };

__device__ __forceinline__ V3 v3(float x, float y, float z) { V3 r{x, y, z}; return r; }
__device__ __forceinline__ V3 vadd(V3 a, V3 b) { return v3(a.x + b.x, a.y + b.y, a.z + b.z); }
__device__ __forceinline__ V3 vscale(float s, V3 a) { return v3(s * a.x, s * a.y, s * a.z); }
__device__ __forceinline__ float vdot(V3 a, V3 b) { return a.x * b.x + a.y * b.y + a.z * b.z; }
__device__ __forceinline__ V3 vcross(V3 a, V3 b) {
  return v3(a.y * b.z - a.z * b.y, a.z * b.x - a.x * b.z, a.x * b.y - a.y * b.x);
}

__device__ __forceinline__ V3 qrot(Q4 q, V3 v) {
  V3 qv = v3(q.x, q.y, q.z);
  V3 t = vscale(2.0f, vcross(qv, v));
  return vadd(vadd(v, vscale(q.w, t)), vcross(qv, t));
}

__device__ __forceinline__ Q4 qmul(Q4 p, Q4 q) {
  V3 pv = v3(p.x, p.y, p.z), qv = v3(q.x, q.y, q.z);
  V3 v = vadd(vadd(vscale(p.w, qv), vscale(q.w, pv)), vcross(pv, qv));
  Q4 r{v.x, v.y, v.z, p.w * q.w - vdot(pv, qv)};
  return r;
}

__device__ __forceinline__ SE3 se3inv(SE3 T) {
  Q4 qi{-T.q.x, -T.q.y, -T.q.z, T.q.w};
  SE3 r;
  r.q = qi;
  r.t = vscale(-1.0f, qrot(qi, T.t));
  return r;
}

__device__ __forceinline__ SE3 se3mul(SE3 A, SE3 B) {
  SE3 r;
  r.t = vadd(A.t, qrot(A.q, B.t));
  r.q = qmul(A.q, B.q);
  return r;
}

// ---- kernel ----------------------------------------------------------------
// One thread per edge. Wave32-private LDS staging:
//   relposes (stride-7 AoS): async coalesced dword stream -> LDS -> stride-7
//     reads (7 coprime 64 banks: conflict-free).
//   output (stride-6 AoS): stride-6 LDS writes (conflict-free) -> 6 coalesced
//     async LDS->global store passes (no VGPR read-back).
__global__ __launch_bounds__(BLOCK) void pgo_err_kernel(
    const int* __restrict__ edges,      // (E,2) int32
    const float* __restrict__ relposes, // (E,7)
    const float* __restrict__ root,     // (1,7)
    const float* __restrict__ nodes,    // (N-1,7)
    float* __restrict__ out,            // (E,6)
    int E) {
  __shared__ float sRel[WAVES * 32 * 7];
  __shared__ float sOut[WAVES * 32 * 6];

  const int tid = threadIdx.x;
  const int lane = tid & 31;
  const int wave = tid >> 5;
  const int wbase = blockIdx.x * BLOCK + wave * 32; // first edge of this wave

  float* wRel = sRel + wave * (32 * 7);
  float* wOut = sOut + wave * (32 * 6);

  // ---- stage 32 relposes (224 dwords) for this wave, coalesced -------------
  const int relBase = wbase * 7;
  const int relTotal = E * 7;
#if ASYNC_LDS
#pragma unroll
  for (int p = 0; p < 7; ++p) {
    int g = relBase + p * 32 + lane;
    if (g >= relTotal) g = relTotal - 1; // clamp tail (no OOB)
    __builtin_amdgcn_global_load_async_to_lds_b32(
        (gas_i32*)(relposes + g),
        (las_i32*)(wRel + p * 32 + lane),
        /*offset=*/0, /*cpol=*/0);
  }
  wait_async0();
  __asm__ volatile("" ::: "memory");
#else
#pragma unroll
  for (int p = 0; p < 7; ++p) {
    int g = relBase + p * 32 + lane;
    if (g >= relTotal) g = relTotal - 1;
    wRel[p * 32 + lane] = relposes[g];
  }
#endif

  const int e = wbase + lane;
  const int esafe = (e < E) ? e : (E - 1);

  // ---- gather the two node poses (L2-resident: 28MB table vs 192MB L2) -----
  const int2 eidx = ((const int2*)edges)[esafe];
  const float* p1 = (eidx.x == 0) ? root : (nodes + (size_t)(eidx.x - 1) * 7);
  const float* p2 = (eidx.y == 0) ? root : (nodes + (size_t)(eidx.y - 1) * 7);

  SE3 n1, n2, r;
  n1.t = v3(p1[0], p1[1], p1[2]);
  n1.q = {p1[3], p1[4], p1[5], p1[6]};
  n2.t = v3(p2[0], p2[1], p2[2]);
  n2.q = {p2[3], p2[4], p2[5], p2[6]};

  const float* rp = wRel + lane * 7; // stride 7: bank-conflict-free
  r.t = v3(rp[0], rp[1], rp[2]);
  r.q = {rp[3], rp[4], rp[5], rp[6]};

  // ---- err = se3_log( inv(r) * inv(n1) * n2 ) ------------------------------
  SE3 M = se3mul(se3mul(se3inv(r), se3inv(n1)), n2);

  V3 qv = v3(M.q.x, M.q.y, M.q.z);
  float qw = M.q.w;
  float n = sqrtf(vdot(qv, qv));
  float qw_safe = (fabsf(qw) < EPS_F) ? 1.0f : qw;
  float angle = 2.0f * atan2f(n, qw);
  float scale = (n < EPS_F) ? (2.0f / qw_safe) : (angle / n);
  V3 phi = vscale(scale, qv);

  float th2 = vdot(phi, phi);
  float th = sqrtf(th2);
  float c;
  if (th < EPS_F) {
    c = 1.0f / 12.0f;
  } else {
    c = 1.0f / th2 - (1.0f + __cosf(th)) / (2.0f * th * __sinf(th));
  }

  // Vinv = I - 0.5*hat(phi) + c*hat(phi)^2 ; rho = Vinv * t
  float x = phi.x, y = phi.y, z = phi.z;
  float xx = x * x, yy = y * y, zz = z * z;
  float xy = x * y, xz = x * z, yz = y * z;
  V3 t = M.t;
  V3 rho;
  rho.x = (1.0f - c * (yy + zz)) * t.x + (0.5f * z + c * xy) * t.y + (-0.5f * y + c * xz) * t.z;
  rho.y = (-0.5f * z + c * xy) * t.x + (1.0f - c * (xx + zz)) * t.y + (0.5f * x + c * yz) * t.z;
  rho.z = (0.5f * y + c * xz) * t.x + (-0.5f * x + c * yz) * t.y + (1.0f - c * (xx + yy)) * t.z;

  // ---- stage output through LDS, then 6 coalesced dword passes -------------
  float* op = wOut + lane * 6; // stride 6: bank-conflict-free
  op[0] = rho.x; op[1] = rho.y; op[2] = rho.z;
  op[3] = phi.x; op[4] = phi.y; op[5] = phi.z;
  // wave-private LDS: same-wave DS ops are in-order, no barrier needed

  const int outBase = wbase * 6;
  const int outTotal = E * 6;
#if ASYNC_ST
  wait_ds0(); // LDS writes visible before TDM-side async stores read LDS
  __asm__ volatile("" ::: "memory");
#pragma unroll
  for (int p = 0; p < 6; ++p) {
    int g = outBase + p * 32 + lane;
    if (g < outTotal) { // EXEC-predicated per-lane async transfer
      __builtin_amdgcn_global_store_async_from_lds_b32(
          (gas_i32*)(out + g),
          (las_i32*)(wOut + p * 32 + lane),
          /*offset=*/0, /*cpol=*/0);
    }
  }
  // ASYNCcnt drains at S_ENDPGM (implicit wait-idle)
#else
#pragma unroll
  for (int p = 0; p < 6; ++p) {
    int g = outBase + p * 32 + lane;
    if (g < outTotal) out[g] = wOut[p * 32 + lane];
  }
#endif
}

// ---- host-side launch ------------------------------------------------------
extern "C" void kernel_launch(void* const* d_in, const int* in_sizes, int n_in,
                              void* d_out, int out_size, void* d_ws, size_t ws_size,
                              hipStream_t stream) {
  const int* edges = (const int*)d_in[0];        // (E,2) int32
  const float* relposes = (const float*)d_in[1]; // (E,7) f32
  const float* root = (const float*)d_in[2];     // (1,7) f32
  const float* nodes = (const float*)d_in[3];    // (N-1,7) f32
  float* out = (float*)d_out;                    // (E,6) f32

  const int E = in_sizes[0] / 2;
  const int blocks = (E + BLOCK - 1) / BLOCK;
  pgo_err_kernel<<<blocks, BLOCK, 0, stream>>>(edges, relposes, root, nodes, out, E);
}